// SchNetInteraction_54039278518702
// MI455X (gfx1250) — compile-verified
//
#include <hip/hip_runtime.h>

// SchNet interaction block for MI455X (gfx1250, wave32, WMMA).
// All GEMMs via V_WMMA_F32_16X16X4_F32 (fp32-exact, matches reference dtype).
//
// Design:
//  - Weights pre-packed into B-fragment order -> each B fragment is ONE
//    coalesced global_load_b64 (was 2x stride-512B b32 loads).
//  - 32 rows per wave (acc[2][8]) -> each B fragment feeds 2 WMMAs.
//  - Hidden activations staged in XOR-swizzled LDS (64KB/block,
//    bank-conflict-free for both C-layout writes and A-fragment b64 reads).
//  - filter_net fused with gather-mul-scatter; W never hits HBM.
//    x (20.5MB) and agg (20.5MB) stay resident in the 192MB L2, so the
//    random gather + 82M atomic f32 adds are on-chip.

typedef __attribute__((ext_vector_type(2))) float v2f;
typedef __attribute__((ext_vector_type(8))) float v8f;

#define NN 40000
#define NE 640000
#define HID 128
#define NGAUSS 50
#define NGPAD 52

__device__ __forceinline__ v8f wmma_f32(v2f a, v2f b, v8f c) {
  // 8 args: (neg_a, A, neg_b, B, c_mod, C, reuse_a, reuse_b)
  return __builtin_amdgcn_wmma_f32_16x16x4_f32(false, a, false, b, (short)0, c,
                                               false, false);
}

__device__ __forceinline__ float silu_f(float z) {
  return z * (1.0f / (1.0f + __expf(-z)));
}

// XOR swizzle: row-major 128-wide, col ^ 4*(row%16).
// A-frag reads (rows r..r+15, even col pair) and C-layout writes both hit all
// 64 LDS banks exactly once per wave -> conflict-free, zero padding waste.
__device__ __forceinline__ int swz(int row, int col) {
  return (row << 7) + (col ^ ((row & 15) << 2));
}

__global__ void zero_kernel(float* __restrict__ p, int n) {
  int i = blockIdx.x * blockDim.x + threadIdx.x;
  if (i < n) p[i] = 0.0f;
}

// Repack W[ksrc x 128] (row-major) into B-fragment order, zero-padded to kpad:
// for kp in [0, kpad/2): pair index kp -> (kstep=kp>>1, half=kp&1),
//   k0 = kstep*4 + half*2;  packed[kp*256 + c*2 + {0,1}] = W[{k0,k0+1}][c]
// Consumer: lane(half,r), tile t, k: b64 at packed[((k>>1)+half)*256 + col*2].
__global__ void pack_w_kernel(const float* __restrict__ w, float* __restrict__ p,
                              int ksrc, int kpad) {
  int i = blockIdx.x * blockDim.x + threadIdx.x;  // pair index
  int npairs = (kpad >> 1) * HID;
  if (i >= npairs) return;
  int kp = i / HID;
  int c  = i % HID;
  int k0 = (kp >> 1) * 4 + (kp & 1) * 2;
  p[2 * i]     = (k0     < ksrc) ? w[(size_t)k0 * HID + c]       : 0.0f;
  p[2 * i + 1] = (k0 + 1 < ksrc) ? w[(size_t)(k0 + 1) * HID + c] : 0.0f;
}

// out[row,:] = silu(in[row,:] @ w1 + b1) @ w2 + b2 (+ res[row,:])
// Block = 128 threads = 4 waves; each wave owns a 32-row strip (2 m-tiles).
__global__ __launch_bounds__(128) void mlp2_kernel(
    const float* __restrict__ in, const float* __restrict__ w1p,
    const float* __restrict__ b1, const float* __restrict__ w2p,
    const float* __restrict__ b2, const float* __restrict__ res,
    float* __restrict__ out, int nrows) {
  __shared__ float hid[128 * HID];  // 64KB, XOR-swizzled
  const int wave = threadIdx.x >> 5;
  const int lane = threadIdx.x & 31;
  const int half = lane >> 4;  // 0: K={0,1}, M 0..7 ; 1: K={2,3}, M 8..15
  const int r    = lane & 15;
  const int wrow = blockIdx.x * 128 + wave * 32;

  // Clamp A-load rows for the tail block (stores are guarded instead;
  // EXEC stays all-ones around every WMMA).
  int r0 = wrow + r;      if (r0 >= nrows) r0 = nrows - 1;
  int r1 = wrow + 16 + r; if (r1 >= nrows) r1 = nrows - 1;
  const float* a0p = &in[(size_t)r0 * HID];
  const float* a1p = &in[(size_t)r1 * HID];

  // ---- GEMM1: [32 x 128] = in_tile @ w1 ----
  v8f acc[2][8] = {};
  for (int k = 0; k < HID; k += 4) {
    const int ka = k + 2 * half;
    const v2f a0 = *(const v2f*)&a0p[ka];
    const v2f a1 = *(const v2f*)&a1p[ka];
    const float* bp = &w1p[((k >> 1) + half) * (2 * HID)];
#pragma unroll
    for (int t = 0; t < 8; ++t) {
      const v2f b = *(const v2f*)&bp[(t * 16 + r) * 2];
      acc[0][t] = wmma_f32(a0, b, acc[0][t]);
      acc[1][t] = wmma_f32(a1, b, acc[1][t]);
    }
  }
  // bias + silu, stage to swizzled LDS (C-layout -> A-layout bridge)
#pragma unroll
  for (int mt = 0; mt < 2; ++mt)
#pragma unroll
    for (int t = 0; t < 8; ++t)
#pragma unroll
      for (int v = 0; v < 8; ++v) {
        const int lrow = wave * 32 + mt * 16 + v + 8 * half;
        const int col  = t * 16 + r;
        hid[swz(lrow, col)] = silu_f(acc[mt][t][v] + b1[col]);
      }
  __syncthreads();

  // ---- GEMM2: [32 x 128] = hid @ w2 ----
  v8f acc2[2][8] = {};
  for (int k = 0; k < HID; k += 4) {
    const int ka = k + 2 * half;
    const v2f a0 = *(const v2f*)&hid[swz(wave * 32 + r, ka)];
    const v2f a1 = *(const v2f*)&hid[swz(wave * 32 + 16 + r, ka)];
    const float* bp = &w2p[((k >> 1) + half) * (2 * HID)];
#pragma unroll
    for (int t = 0; t < 8; ++t) {
      const v2f b = *(const v2f*)&bp[(t * 16 + r) * 2];
      acc2[0][t] = wmma_f32(a0, b, acc2[0][t]);
      acc2[1][t] = wmma_f32(a1, b, acc2[1][t]);
    }
  }
#pragma unroll
  for (int mt = 0; mt < 2; ++mt)
#pragma unroll
    for (int t = 0; t < 8; ++t)
#pragma unroll
      for (int v = 0; v < 8; ++v) {
        const int row = wrow + mt * 16 + v + 8 * half;
        const int col = t * 16 + r;
        if (row < nrows) {
          const size_t idx = (size_t)row * HID + col;
          float val = acc2[mt][t][v] + b2[col];
          if (res) val += res[idx];
          out[idx] = val;
        }
      }
}

// Fused filter_net + gather-multiply-scatter:
//   W = silu(dist @ fw1 + fb1) @ fw2 + fb2   (never materialized in HBM)
//   agg[dst] += W * x[src]                   (atomics land in L2)
__global__ __launch_bounds__(128) void edge_kernel(
    const float* __restrict__ dist, const long long* __restrict__ ei,
    const float* __restrict__ fw1p, const float* __restrict__ fb1,
    const float* __restrict__ fw2p, const float* __restrict__ fb2,
    const float* __restrict__ x, float* __restrict__ agg) {
  __shared__ float hid[128 * HID];  // 64KB, XOR-swizzled
  const int wave = threadIdx.x >> 5;
  const int lane = threadIdx.x & 31;
  const int half = lane >> 4;
  const int r    = lane & 15;
  const int e0   = blockIdx.x * 128 + wave * 32;  // 640000/128 = 5000 exact

  // ---- GEMM1: [32 x 128] = dist_tile[32 x 52] @ fw1p[52 x 128] ----
  // fw1p rows 50,51 are zero (pack kernel); a pads are zeroed here, so the
  // K=50->52 padding contributes exactly 0 (no garbage/NaN can enter).
  v8f acc[2][8] = {};
  for (int k = 0; k < NGPAD; k += 4) {
    const int ka = k + 2 * half;
    v2f a0 = {0.0f, 0.0f}, a1 = {0.0f, 0.0f};
    if (ka < NGAUSS) {  // pairs are (48,49) in-bounds or (50,51) fully padded
      a0 = *(const v2f*)&dist[(size_t)(e0 + r) * NGAUSS + ka];
      a1 = *(const v2f*)&dist[(size_t)(e0 + 16 + r) * NGAUSS + ka];
    }
    const float* bp = &fw1p[((k >> 1) + half) * (2 * HID)];
#pragma unroll
    for (int t = 0; t < 8; ++t) {
      const v2f b = *(const v2f*)&bp[(t * 16 + r) * 2];
      acc[0][t] = wmma_f32(a0, b, acc[0][t]);
      acc[1][t] = wmma_f32(a1, b, acc[1][t]);
    }
  }
#pragma unroll
  for (int mt = 0; mt < 2; ++mt)
#pragma unroll
    for (int t = 0; t < 8; ++t)
#pragma unroll
      for (int v = 0; v < 8; ++v) {
        const int lrow = wave * 32 + mt * 16 + v + 8 * half;
        const int col  = t * 16 + r;
        hid[swz(lrow, col)] = silu_f(acc[mt][t][v] + fb1[col]);
      }
  __syncthreads();

  // ---- GEMM2: W[32 x 128] = hid @ fw2 ----
  v8f acc2[2][8] = {};
  for (int k = 0; k < HID; k += 4) {
    const int ka = k + 2 * half;
    const v2f a0 = *(const v2f*)&hid[swz(wave * 32 + r, ka)];
    const v2f a1 = *(const v2f*)&hid[swz(wave * 32 + 16 + r, ka)];
    const float* bp = &fw2p[((k >> 1) + half) * (2 * HID)];
#pragma unroll
    for (int t = 0; t < 8; ++t) {
      const v2f b = *(const v2f*)&bp[(t * 16 + r) * 2];
      acc2[0][t] = wmma_f32(a0, b, acc2[0][t]);
      acc2[1][t] = wmma_f32(a1, b, acc2[1][t]);
    }
  }

  // ---- gather x[src], multiply, scatter-add into agg[dst] ----
#pragma unroll
  for (int mt = 0; mt < 2; ++mt)
#pragma unroll
    for (int v = 0; v < 8; ++v) {
      const int e = e0 + mt * 16 + v + 8 * half;
      const long long s = ei[e];       // edge_index[0][e] (int64)
      const long long d = ei[NE + e];  // edge_index[1][e]
      const float* xr = &x[(size_t)s * HID];
      float*       ar = &agg[(size_t)d * HID];
#pragma unroll
      for (int t = 0; t < 8; ++t) {
        const int col = t * 16 + r;
        const float msg = (acc2[mt][t][v] + fb2[col]) * xr[col];
        unsafeAtomicAdd(&ar[col], msg);  // global_atomic_add_f32
      }
    }
}

extern "C" void kernel_launch(void* const* d_in, const int* in_sizes, int n_in,
                              void* d_out, int out_size, void* d_ws,
                              size_t ws_size, hipStream_t stream) {
  const float*     h    = (const float*)d_in[0];
  const long long* ei   = (const long long*)d_in[1];  // int64 in reference
  const float*     dist = (const float*)d_in[2];
  const float* fw1 = (const float*)d_in[3];
  const float* fb1 = (const float*)d_in[4];
  const float* fw2 = (const float*)d_in[5];
  const float* fb2 = (const float*)d_in[6];
  const float* aw1 = (const float*)d_in[7];
  const float* ab1 = (const float*)d_in[8];
  const float* aw2 = (const float*)d_in[9];
  const float* ab2 = (const float*)d_in[10];
  const float* ow1 = (const float*)d_in[11];
  const float* ob1 = (const float*)d_in[12];
  const float* ow2 = (const float*)d_in[13];
  const float* ob2 = (const float*)d_in[14];
  float* out = (float*)d_out;

  // Workspace layout (floats): x | agg | packed weights (~41.3 MB total)
  float* x    = (float*)d_ws;
  float* agg  = x + (size_t)NN * HID;
  float* fw1p = agg + (size_t)NN * HID;
  float* fw2p = fw1p + NGPAD * HID;
  float* aw1p = fw2p + HID * HID;
  float* aw2p = aw1p + HID * HID;
  float* ow1p = aw2p + HID * HID;
  float* ow2p = ow1p + HID * HID;

  const int nelem = NN * HID;
  zero_kernel<<<(nelem + 255) / 256, 256, 0, stream>>>(agg, nelem);

  const int np50  = (NGPAD / 2) * HID;  // 3328 pairs
  const int np128 = (HID / 2) * HID;    // 8192 pairs
  pack_w_kernel<<<(np50 + 255) / 256, 256, 0, stream>>>(fw1, fw1p, NGAUSS, NGPAD);
  pack_w_kernel<<<(np128 + 255) / 256, 256, 0, stream>>>(fw2, fw2p, HID, HID);
  pack_w_kernel<<<(np128 + 255) / 256, 256, 0, stream>>>(aw1, aw1p, HID, HID);
  pack_w_kernel<<<(np128 + 255) / 256, 256, 0, stream>>>(aw2, aw2p, HID, HID);
  pack_w_kernel<<<(np128 + 255) / 256, 256, 0, stream>>>(ow1, ow1p, HID, HID);
  pack_w_kernel<<<(np128 + 255) / 256, 256, 0, stream>>>(ow2, ow2p, HID, HID);

  // atom_net: x = silu(h @ aw1 + ab1) @ aw2 + ab2
  mlp2_kernel<<<(NN + 127) / 128, 128, 0, stream>>>(h, aw1p, ab1, aw2p, ab2,
                                                    nullptr, x, NN);
  // fused filter_net + message scatter
  edge_kernel<<<NE / 128, 128, 0, stream>>>(dist, ei, fw1p, fb1, fw2p, fb2, x,
                                            agg);
  // output_net + residual: out = h + silu(agg @ ow1 + ob1) @ ow2 + ob2
  mlp2_kernel<<<(NN + 127) / 128, 128, 0, stream>>>(agg, ow1p, ob1, ow2p, ob2,
                                                    h, out, NN);
}